// DifferentiableBiquad_28810640621773
// MI455X (gfx1250) — compile-verified
//
#include <hip/hip_runtime.h>
#include <math.h>

// ---------------------------------------------------------------------------
// Biquad cascade (4 filters) over B=32 rows, L=524288 samples.
// Exact chunked parallel-scan formulation of the 8-state linear system
//   s[n] = A s[n-1] + b x[n],  y[n] = c.s[n-1] + d x[n]
// Pass1: per-chunk zero-entry final states (WMMA blocked Toeplitz form)
// Scan : two-level entry-state scan with constant transition M = A^CHUNK
// Pass2: per-chunk output with correct entry state (WMMA)
// ---------------------------------------------------------------------------

#define B_ROWS   32
#define L_LEN    524288
#define CHUNK    1024
#define NC       (L_LEN / CHUNK)      // 512 chunks per row
#define WPR      (NC / 16)            // 32 waves per row (16 chunk-cols/wave)
#define BLK_IT   (CHUNK / 16)         // 64 sixteen-sample blocks per chunk
#define GROUPS   16                   // scan groups per row
#define CPG      (NC / GROUPS)        // 32 chunks per group

// workspace layout (in floats)
#define T_OFF    0                    // [16][16] Toeplitz of impulse response
#define P_OFF    256                  // [16][8]  rows c^T A^i
#define Q_OFF    384                  // [16][16] cols A^{15-j} b (rows 8..15 = 0)
#define A16_OFF  640                  // [16][8]  A^16 (rows 8..15 = 0)
#define M_OFF    768                  // [8][8]   A^CHUNK
#define M32_OFF  832                  // [8][8]   A^(CHUNK*CPG)
#define F_OFF    1024                 // [B][NC][8] zero-entry chunk finals
#define E_OFF    (F_OFF + B_ROWS*NC*8)      // [B][NC][8] chunk entry states
#define G_OFF    (E_OFF + B_ROWS*NC*8)      // [B][GROUPS][8] group totals
#define GE_OFF   (G_OFF + B_ROWS*GROUPS*8)  // [B][GROUPS][8] group entries

typedef __attribute__((ext_vector_type(2))) float v2f;
typedef __attribute__((ext_vector_type(8))) float v8f;

__device__ __forceinline__ v8f wmma4(v2f a, v2f b, v8f c) {
  // D(16x16) = A(16x4) * B(4x16) + C   -- V_WMMA_F32_16X16X4_F32
  return __builtin_amdgcn_wmma_f32_16x16x4_f32(
      false, a, false, b, (short)0, c, false, false);
}

// A-operand layout (16x4 f32): VGPR v, lane l -> M = l%16, K = 4j + v + 2*(l/16)
__device__ __forceinline__ v2f loadA(const float* mat, int ncols, int j, int l) {
  int m = l & 15;
  int kb = 4 * j + ((l >> 4) << 1);
  v2f a;
  a.x = mat[m * ncols + kb];
  a.y = mat[m * ncols + kb + 1];
  return a;
}

// ---------------------------------------------------------------------------
// Setup: single thread builds all constant matrices from filter params.
// ---------------------------------------------------------------------------
__device__ __forceinline__ float cascade_step(const float* B0, const float* B1,
                                              const float* B2, const float* a1,
                                              const float* a2, const float* s,
                                              float x, float* sn) {
  float u = x;
#pragma unroll
  for (int f = 0; f < 4; f++) {
    float s1 = s[2 * f], s2 = s[2 * f + 1];
    float yv = B0[f] * u + s1;
    sn[2 * f]     = B1[f] * u - a1[f] * yv + s2;
    sn[2 * f + 1] = B2[f] * u - a2[f] * yv;
    u = yv;
  }
  return u;
}

__device__ __forceinline__ void mm8(float* dst, const float* a, const float* b) {
  for (int i = 0; i < 8; i++)
    for (int j = 0; j < 8; j++) {
      float acc = 0.f;
      for (int k = 0; k < 8; k++) acc += a[i * 8 + k] * b[k * 8 + j];
      dst[i * 8 + j] = acc;
    }
}

__global__ void biq_setup(const float* lr, const float* ra, const float* b0,
                          const float* b1, const float* b2, float* ws) {
  if (threadIdx.x != 0 || blockIdx.x != 0) return;
  float a1[4], a2[4], B0[4], B1[4], B2[4];
  for (int f = 0; f < 4; f++) {
    float r = (1.f / (1.f + expf(-lr[f]))) * 0.999f;
    float ang = (1.f / (1.f + expf(-ra[f]))) * 3.14159265358979323846f;
    a1[f] = -2.f * r * cosf(ang);
    a2[f] = r * r;
    B0[f] = b0[f]; B1[f] = b1[f]; B2[f] = b2[f];
  }
  // build A (8x8), b, c, d by basis evaluation of the step function
  float A[64], bv[8], cv[8], d, s[8], sn[8];
  for (int i = 0; i < 8; i++) {
    for (int k = 0; k < 8; k++) s[k] = (k == i) ? 1.f : 0.f;
    cv[i] = cascade_step(B0, B1, B2, a1, a2, s, 0.f, sn);
    for (int r = 0; r < 8; r++) A[r * 8 + i] = sn[r];
  }
  for (int k = 0; k < 8; k++) s[k] = 0.f;
  d = cascade_step(B0, B1, B2, a1, a2, s, 1.f, sn);
  for (int r = 0; r < 8; r++) bv[r] = sn[r];
  // impulse response h[0..15]
  float h[16], v[8];
  h[0] = d;
  for (int r = 0; r < 8; r++) v[r] = bv[r];
  for (int k = 1; k < 16; k++) {
    float acc = 0.f;
    for (int r = 0; r < 8; r++) acc += cv[r] * v[r];
    h[k] = acc;
    float nv[8];
    for (int i = 0; i < 8; i++) {
      float t = 0.f;
      for (int j = 0; j < 8; j++) t += A[i * 8 + j] * v[j];
      nv[i] = t;
    }
    for (int r = 0; r < 8; r++) v[r] = nv[r];
  }
  // T: lower-triangular Toeplitz
  for (int i = 0; i < 16; i++)
    for (int j = 0; j < 16; j++)
      ws[T_OFF + i * 16 + j] = (j <= i) ? h[i - j] : 0.f;
  // P: row i = c^T A^i
  float p[8];
  for (int r = 0; r < 8; r++) p[r] = cv[r];
  for (int i = 0; i < 16; i++) {
    for (int k = 0; k < 8; k++) ws[P_OFF + i * 8 + k] = p[k];
    float np[8];
    for (int j = 0; j < 8; j++) {
      float t = 0.f;
      for (int k = 0; k < 8; k++) t += p[k] * A[k * 8 + j];
      np[j] = t;
    }
    for (int r = 0; r < 8; r++) p[r] = np[r];
  }
  // Q: col j = A^{15-j} b (padded to 16 rows)
  float w[8];
  for (int r = 0; r < 8; r++) w[r] = bv[r];
  for (int j = 15; j >= 0; j--) {
    for (int r = 0; r < 8; r++) ws[Q_OFF + r * 16 + j] = w[r];
    float nw[8];
    for (int i = 0; i < 8; i++) {
      float t = 0.f;
      for (int k = 0; k < 8; k++) t += A[i * 8 + k] * w[k];
      nw[i] = t;
    }
    for (int r = 0; r < 8; r++) w[r] = nw[r];
  }
  for (int r = 8; r < 16; r++)
    for (int j = 0; j < 16; j++) ws[Q_OFF + r * 16 + j] = 0.f;
  // matrix powers: A16, M = A^CHUNK, M32 = M^CPG (repeated squaring)
  float Ac[64], Tm[64];
  for (int i = 0; i < 64; i++) Ac[i] = A[i];
  for (int t = 0; t < 4; t++) { mm8(Tm, Ac, Ac); for (int i = 0; i < 64; i++) Ac[i] = Tm[i]; } // A^16
  for (int r = 0; r < 8; r++)
    for (int k = 0; k < 8; k++) ws[A16_OFF + r * 8 + k] = Ac[r * 8 + k];
  for (int r = 8; r < 16; r++)
    for (int k = 0; k < 8; k++) ws[A16_OFF + r * 8 + k] = 0.f;
  for (int t = 0; t < 6; t++) { mm8(Tm, Ac, Ac); for (int i = 0; i < 64; i++) Ac[i] = Tm[i]; } // A^1024
  for (int i = 0; i < 64; i++) ws[M_OFF + i] = Ac[i];
  for (int t = 0; t < 5; t++) { mm8(Tm, Ac, Ac); for (int i = 0; i < 64; i++) Ac[i] = Tm[i]; } // A^32768
  for (int i = 0; i < 64; i++) ws[M32_OFF + i] = Ac[i];
}

// ---------------------------------------------------------------------------
// Pass 1: per-chunk zero-entry final state via WMMA (Snew = A16*S + Q*X)
// one wave = 16 chunk-columns of one row
// ---------------------------------------------------------------------------
__global__ __launch_bounds__(32) void biq_pass1(const float* __restrict__ x,
                                                float* __restrict__ ws) {
  __shared__ float Sbuf[128];  // 8 rows x 16 cols
  const int wave = blockIdx.x;
  const int row = wave / WPR;
  const int cb = (wave % WPR) * 16;
  const int l = threadIdx.x;
  const int col = l & 15, hh = l >> 4;

  v2f Qa[4], A16a[2];
#pragma unroll
  for (int j = 0; j < 4; j++) Qa[j] = loadA(ws + Q_OFF, 16, j, l);
#pragma unroll
  for (int j = 0; j < 2; j++) A16a[j] = loadA(ws + A16_OFF, 8, j, l);

  if (l < 16) {
#pragma unroll
    for (int r = 0; r < 8; r++) Sbuf[r * 16 + l] = 0.f;
  }
  __syncthreads();

  const float* xcol = x + (size_t)row * L_LEN + (size_t)(cb + col) * CHUNK + 2 * hh;

  for (int blk = 0; blk < BLK_IT; blk++) {
    const int sb = blk * 16;
    v2f Bs0, Bs1;
    Bs0.x = Sbuf[(0 + 2 * hh) * 16 + col];
    Bs0.y = Sbuf[(1 + 2 * hh) * 16 + col];
    Bs1.x = Sbuf[(4 + 2 * hh) * 16 + col];
    Bs1.y = Sbuf[(5 + 2 * hh) * 16 + col];
    v8f Sacc = {};
#pragma unroll
    for (int j = 0; j < 4; j++) {
      const float* bp = xcol + sb + 4 * j;
      v2f Bx; Bx.x = bp[0]; Bx.y = bp[1];
      Sacc = wmma4(Qa[j], Bx, Sacc);
    }
    Sacc = wmma4(A16a[0], Bs0, Sacc);
    Sacc = wmma4(A16a[1], Bs1, Sacc);
    __syncthreads();
    if (l < 16) {
#pragma unroll
      for (int r = 0; r < 8; r++) Sbuf[r * 16 + l] = Sacc[r];
    }
    __syncthreads();
  }
  if (l < 16) {
    float* F = ws + F_OFF + ((size_t)row * NC + cb + l) * 8;
#pragma unroll
    for (int r = 0; r < 8; r++) F[r] = Sbuf[r * 16 + l];
  }
}

// ---------------------------------------------------------------------------
// Scan: entry states. E_{c+1} = M*E_c + F_c, constant M.
// ---------------------------------------------------------------------------
__device__ __forceinline__ void madd8(const float* M, float* E, const float* F) {
  float t[8];
  for (int i = 0; i < 8; i++) {
    float acc = F[i];
    for (int k = 0; k < 8; k++) acc += M[i * 8 + k] * E[k];
    t[i] = acc;
  }
  for (int i = 0; i < 8; i++) E[i] = t[i];
}

__global__ void biq_scan_groups(float* ws) {  // 512 threads: zero-entry group totals
  int gid = blockIdx.x * blockDim.x + threadIdx.x;
  if (gid >= B_ROWS * GROUPS) return;
  int row = gid / GROUPS, g = gid % GROUPS;
  const float* M = ws + M_OFF;
  float E[8] = {0, 0, 0, 0, 0, 0, 0, 0};
  for (int i = 0; i < CPG; i++) {
    const float* F = ws + F_OFF + ((size_t)row * NC + g * CPG + i) * 8;
    madd8(M, E, F);
  }
  float* G = ws + G_OFF + (size_t)gid * 8;
  for (int r = 0; r < 8; r++) G[r] = E[r];
}

__global__ void biq_scan_rows(float* ws) {  // 32 threads: group entry states
  int row = blockIdx.x * blockDim.x + threadIdx.x;
  if (row >= B_ROWS) return;
  const float* M32 = ws + M32_OFF;
  float E[8] = {0, 0, 0, 0, 0, 0, 0, 0};
  for (int g = 0; g < GROUPS; g++) {
    float* GE = ws + GE_OFF + ((size_t)row * GROUPS + g) * 8;
    for (int r = 0; r < 8; r++) GE[r] = E[r];
    const float* G = ws + G_OFF + ((size_t)row * GROUPS + g) * 8;
    madd8(M32, E, G);
  }
}

__global__ void biq_scan_chunks(float* ws) {  // 512 threads: per-chunk entries
  int gid = blockIdx.x * blockDim.x + threadIdx.x;
  if (gid >= B_ROWS * GROUPS) return;
  int row = gid / GROUPS, g = gid % GROUPS;
  const float* M = ws + M_OFF;
  float E[8];
  const float* GE = ws + GE_OFF + (size_t)gid * 8;
  for (int r = 0; r < 8; r++) E[r] = GE[r];
  for (int i = 0; i < CPG; i++) {
    size_t c = (size_t)row * NC + g * CPG + i;
    float* Eo = ws + E_OFF + c * 8;
    for (int r = 0; r < 8; r++) Eo[r] = E[r];
    madd8(M, E, ws + F_OFF + c * 8);
  }
}

// ---------------------------------------------------------------------------
// Pass 2: outputs. Y = T*X + P*S ; Snew = A16*S + Q*X   (12 WMMA / block)
// ---------------------------------------------------------------------------
__global__ __launch_bounds__(32) void biq_pass2(const float* __restrict__ x,
                                                const float* __restrict__ ws,
                                                float* __restrict__ y) {
  __shared__ float Sbuf[128];
  const int wave = blockIdx.x;
  const int row = wave / WPR;
  const int cb = (wave % WPR) * 16;
  const int l = threadIdx.x;
  const int col = l & 15, hh = l >> 4;

  v2f Ta[4], Pa[2], Qa[4], A16a[2];
#pragma unroll
  for (int j = 0; j < 4; j++) Ta[j] = loadA(ws + T_OFF, 16, j, l);
#pragma unroll
  for (int j = 0; j < 2; j++) Pa[j] = loadA(ws + P_OFF, 8, j, l);
#pragma unroll
  for (int j = 0; j < 4; j++) Qa[j] = loadA(ws + Q_OFF, 16, j, l);
#pragma unroll
  for (int j = 0; j < 2; j++) A16a[j] = loadA(ws + A16_OFF, 8, j, l);

  if (l < 16) {
    const float* E = ws + E_OFF + ((size_t)row * NC + cb + l) * 8;
#pragma unroll
    for (int r = 0; r < 8; r++) Sbuf[r * 16 + l] = E[r];
  }
  __syncthreads();

  const float* xcol = x + (size_t)row * L_LEN + (size_t)(cb + col) * CHUNK + 2 * hh;
  float* ycol = y + (size_t)row * L_LEN + (size_t)(cb + col) * CHUNK + 8 * hh;

  for (int blk = 0; blk < BLK_IT; blk++) {
    const int sb = blk * 16;
    v2f Bs0, Bs1;
    Bs0.x = Sbuf[(0 + 2 * hh) * 16 + col];
    Bs0.y = Sbuf[(1 + 2 * hh) * 16 + col];
    Bs1.x = Sbuf[(4 + 2 * hh) * 16 + col];
    Bs1.y = Sbuf[(5 + 2 * hh) * 16 + col];
    v8f Yacc = {};
    v8f Sacc = {};
#pragma unroll
    for (int j = 0; j < 4; j++) {
      const float* bp = xcol + sb + 4 * j;
      v2f Bx; Bx.x = bp[0]; Bx.y = bp[1];
      Yacc = wmma4(Ta[j], Bx, Yacc);
      Sacc = wmma4(Qa[j], Bx, Sacc);
    }
    Yacc = wmma4(Pa[0], Bs0, Yacc);
    Yacc = wmma4(Pa[1], Bs1, Yacc);
    Sacc = wmma4(A16a[0], Bs0, Sacc);
    Sacc = wmma4(A16a[1], Bs1, Sacc);
    // D layout: VGPR r, lane l -> sample M = r + 8*(l/16), col N = l%16
    float* yp = ycol + sb;
    float4 lo = make_float4(Yacc[0], Yacc[1], Yacc[2], Yacc[3]);
    float4 hi = make_float4(Yacc[4], Yacc[5], Yacc[6], Yacc[7]);
    reinterpret_cast<float4*>(yp)[0] = lo;
    reinterpret_cast<float4*>(yp + 4)[0] = hi;
    __syncthreads();
    if (l < 16) {
#pragma unroll
      for (int r = 0; r < 8; r++) Sbuf[r * 16 + l] = Sacc[r];
    }
    __syncthreads();
  }
}

// ---------------------------------------------------------------------------
extern "C" void kernel_launch(void* const* d_in, const int* in_sizes, int n_in,
                              void* d_out, int out_size, void* d_ws, size_t ws_size,
                              hipStream_t stream) {
  const float* x  = (const float*)d_in[0];
  const float* lr = (const float*)d_in[1];
  const float* ra = (const float*)d_in[2];
  const float* b0 = (const float*)d_in[3];
  const float* b1 = (const float*)d_in[4];
  const float* b2 = (const float*)d_in[5];
  float* y  = (float*)d_out;
  float* ws = (float*)d_ws;

  biq_setup<<<1, 1, 0, stream>>>(lr, ra, b0, b1, b2, ws);
  biq_pass1<<<B_ROWS * WPR, 32, 0, stream>>>(x, ws);
  biq_scan_groups<<<(B_ROWS * GROUPS + 63) / 64, 64, 0, stream>>>(ws);
  biq_scan_rows<<<1, 32, 0, stream>>>(ws);
  biq_scan_chunks<<<(B_ROWS * GROUPS + 63) / 64, 64, 0, stream>>>(ws);
  biq_pass2<<<B_ROWS * WPR, 32, 0, stream>>>(x, ws, y);
}